// NeuralODEgnn_47725676593559
// MI455X (gfx1250) — compile-verified
//
#include <hip/hip_runtime.h>
#include <math.h>

#define Bsz 16
#define Nsz 512
#define Hsz 128
#define NSTEPS 8
#define DTc 0.01f

typedef __attribute__((ext_vector_type(2))) float v2f;
typedef __attribute__((ext_vector_type(8))) float v8f;

// ---------------- one-time weight reductions ----------------
// K0sT[j*H + h] = sum_n k0[h, n, j]   (k0: [H, N, N])
__global__ void reduce_k0(const float* __restrict__ k0, float* __restrict__ K0sT) {
    int idx = blockIdx.x * blockDim.x + threadIdx.x;   // H*N = 65536 threads
    int h = idx >> 9, j = idx & (Nsz - 1);
    const float* p = k0 + (size_t)h * (Nsz * Nsz) + j;
    float acc = 0.f;
#pragma unroll 8
    for (int n = 0; n < Nsz; ++n) acc += p[n * Nsz];
    K0sT[j * Hsz + h] = acc;
}

// K1sT[j*H + h] = sum_n k1[h, n, j]   (k1: [H, N, H])
__global__ void reduce_k1(const float* __restrict__ k1, float* __restrict__ K1sT) {
    int idx = blockIdx.x * blockDim.x + threadIdx.x;   // H*H = 16384 threads
    int h = idx >> 7, j = idx & (Hsz - 1);
    const float* p = k1 + (size_t)h * (Nsz * Hsz) + j;
    float acc = 0.f;
#pragma unroll 8
    for (int n = 0; n < Nsz; ++n) acc += p[n * Hsz];
    K1sT[j * Hsz + h] = acc;
}

// ---------------- tiny GEMMs: out[b,h] = sum_j x[b,j] * WT[j,h] ----------------
__global__ void small_gemm(const float* __restrict__ x, const float* __restrict__ WT,
                           float* __restrict__ out, int K) {
    int idx = blockIdx.x * blockDim.x + threadIdx.x;   // B*H = 2048 threads
    int b = idx >> 7, h = idx & (Hsz - 1);
    float acc = 0.f;
    for (int j = 0; j < K; ++j) acc += x[b * K + j] * WT[j * Hsz + h];
    out[idx] = acc;
}

// ---------------- the big WMMA kernel ----------------
// rhs[b,m] = bias[b,m] - sum_n adj[b,m,n] * sin( sum_j s2[b,j]*k2[m,n,j] )
// one block per m; 8 waves; each wave: 4 n-tiles of 16, K=128 via 32x wmma f32 16x16x4
__global__ void __launch_bounds__(256) rhs_big(
    const float* __restrict__ s2,    // [B, H]
    const float* __restrict__ k2,    // [N, N, H]
    const float* __restrict__ adj,   // [B, N, N]
    const float* __restrict__ bias,  // [B, N]
    float* __restrict__ rhs)         // [B, N]
{
    const int m    = blockIdx.x;
    const int tid  = threadIdx.x;
    const int wave = tid >> 5;
    const int lane = tid & 31;
    const int half = lane >> 4;
    const int l16  = lane & 15;

    __shared__ float agg_sh[Bsz];
    if (tid < Bsz) agg_sh[tid] = 0.f;
    __syncthreads();

    // Preload A fragments: A is 16x4 per step; lane = M (=batch b),
    // lanes 0-15 hold K = 4t..4t+1, lanes 16-31 hold K = 4t+2..4t+3.
    v2f a[32];
#pragma unroll
    for (int t = 0; t < 32; ++t)
        a[t] = *reinterpret_cast<const v2f*>(s2 + l16 * Hsz + 4 * t + 2 * half);

    float agg_acc[8] = {0, 0, 0, 0, 0, 0, 0, 0};

    for (int tt = 0; tt < 4; ++tt) {
        const int tile = wave + 8 * tt;           // n-tile index 0..31
        const int n0   = tile * 16;
        // B is 4x16 (KxN): lane = N column, k2[m, n0+l16, :] is contiguous.
        const float* pb0 = k2 + ((size_t)m * Nsz + n0 + l16) * Hsz + 2 * half;

        v8f c = {0.f, 0.f, 0.f, 0.f, 0.f, 0.f, 0.f, 0.f};
#pragma unroll
        for (int t = 0; t < 32; ++t) {
            v2f bfrag = *reinterpret_cast<const v2f*>(pb0 + 4 * t);
            c = __builtin_amdgcn_wmma_f32_16x16x4_f32(
                    false, a[t], false, bfrag, (short)0, c, false, false);
        }

        // D layout: vgpr r holds (b = r + 8*half, n = n0 + l16)
        const int n = n0 + l16;
        union { v8f v; float f[8]; } cu; cu.v = c;
#pragma unroll
        for (int r = 0; r < 8; ++r) {
            int bidx = r + 8 * half;
            float av = adj[(size_t)bidx * (Nsz * Nsz) + (size_t)m * Nsz + n];
            agg_acc[r] += sinf(cu.f[r]) * av;
        }
    }

    // reduce over the 16 lanes of each half (the n direction)
#pragma unroll
    for (int r = 0; r < 8; ++r) {
        float v = agg_acc[r];
        v += __shfl_xor(v, 1, 32);
        v += __shfl_xor(v, 2, 32);
        v += __shfl_xor(v, 4, 32);
        v += __shfl_xor(v, 8, 32);
        agg_acc[r] = v;
    }
    if (l16 == 0) {
#pragma unroll
        for (int r = 0; r < 8; ++r)
            atomicAdd(&agg_sh[r + 8 * half], agg_acc[r]);
    }
    __syncthreads();

    if (tid < Bsz)
        rhs[tid * Nsz + m] = bias[tid * Nsz + m] - agg_sh[tid];
}

// ---------------- elementwise helpers ----------------
__global__ void copy_k(const float* __restrict__ a, float* __restrict__ b) {
    int i = blockIdx.x * blockDim.x + threadIdx.x;
    b[i] = a[i];
}

__global__ void axpy_k(const float* __restrict__ y, const float* __restrict__ r,
                       float c, float* __restrict__ out) {
    int i = blockIdx.x * blockDim.x + threadIdx.x;
    out[i] = y[i] + c * r[i];
}

__global__ void rk4_combine(float* __restrict__ y,
                            const float* __restrict__ r1, const float* __restrict__ r2,
                            const float* __restrict__ r3, const float* __restrict__ r4,
                            float* __restrict__ out) {
    int i = blockIdx.x * blockDim.x + threadIdx.x;
    float yn = y[i] + (DTc / 6.0f) * (r1[i] + 2.0f * r2[i] + 2.0f * r3[i] + r4[i]);
    y[i] = yn;
    out[i] = yn;
}

extern "C" void kernel_launch(void* const* d_in, const int* in_sizes, int n_in,
                              void* d_out, int out_size, void* d_ws, size_t ws_size,
                              hipStream_t stream) {
    (void)in_sizes; (void)n_in; (void)out_size; (void)ws_size;

    const float* y0   = (const float*)d_in[0];   // [B, N]
    const float* bias = (const float*)d_in[1];   // [B, N, 1]
    const float* adj  = (const float*)d_in[2];   // [B, N, N]
    const float* k0   = (const float*)d_in[3];   // [H, N, N]
    const float* k1   = (const float*)d_in[4];   // [H, N, H]
    const float* k2   = (const float*)d_in[5];   // [N, N, H]
    float* out = (float*)d_out;                  // [NSTEPS, B, N]

    float* ws   = (float*)d_ws;
    float* K0sT = ws; ws += Hsz * Nsz;   // 65536
    float* K1sT = ws; ws += Hsz * Hsz;   // 16384
    float* s1   = ws; ws += Bsz * Hsz;   // 2048
    float* s2   = ws; ws += Bsz * Hsz;   // 2048
    float* ycur = ws; ws += Bsz * Nsz;   // 8192
    float* ytmp = ws; ws += Bsz * Nsz;   // 8192
    float* rst[4];
    for (int i = 0; i < 4; ++i) { rst[i] = ws; ws += Bsz * Nsz; }

    reduce_k0<<<(Hsz * Nsz) / 256, 256, 0, stream>>>(k0, K0sT);
    reduce_k1<<<(Hsz * Hsz) / 256, 256, 0, stream>>>(k1, K1sT);
    copy_k<<<(Bsz * Nsz) / 256, 256, 0, stream>>>(y0, ycur);

    const float cf[3] = {0.5f * DTc, 0.5f * DTc, DTc};
    for (int step = 0; step < NSTEPS; ++step) {
        for (int s = 0; s < 4; ++s) {
            const float* ye = (s == 0) ? ycur : ytmp;
            small_gemm<<<(Bsz * Hsz) / 256, 256, 0, stream>>>(ye, K0sT, s1, Nsz);
            small_gemm<<<(Bsz * Hsz) / 256, 256, 0, stream>>>(s1, K1sT, s2, Hsz);
            rhs_big<<<Nsz, 256, 0, stream>>>(s2, k2, adj, bias, rst[s]);
            if (s < 3)
                axpy_k<<<(Bsz * Nsz) / 256, 256, 0, stream>>>(ycur, rst[s], cf[s], ytmp);
        }
        rk4_combine<<<(Bsz * Nsz) / 256, 256, 0, stream>>>(
            ycur, rst[0], rst[1], rst[2], rst[3], out + step * Bsz * Nsz);
    }
}